// modeler_44220983279800
// MI455X (gfx1250) — compile-verified
//
#include <hip/hip_runtime.h>
#include <hip/hip_bf16.h>

typedef float v2f __attribute__((ext_vector_type(2)));
typedef float v8f __attribute__((ext_vector_type(8)));

// ---------------------------------------------------------------------------
// C[M,K] = act( X[M,Din] @ W[Din,K] (+ bias) ), K = NT*16, fp32 WMMA 16x16x4.
// One wave computes a 16-row strip across all NT column tiles.
// A layout (16x4 f32): lanes 0-15 -> M=lane, vgpr0=K+0, vgpr1=K+1
//                      lanes 16-31 -> M=lane-16, vgpr0=K+2, vgpr1=K+3
// B layout (4x16 f32): mirrored (lane&15 = N, half selects K+0/1 vs K+2/3)
// C layout: vgpr i: lanes 0-15 -> M=i, N=lane; lanes 16-31 -> M=8+i, N=lane-16
// ---------------------------------------------------------------------------
template<int NT, int ACT>
__global__ void gemm_wmma_kernel(const float* __restrict__ X,
                                 const float* __restrict__ W,
                                 const float* __restrict__ bias,
                                 float* __restrict__ C,
                                 int M, int Din) {
    const int K = NT * 16;
    const int wib  = threadIdx.x >> 5;
    const int lane = threadIdx.x & 31;
    const int tM = blockIdx.x * (blockDim.x >> 5) + wib;
    if (tM * 16 >= M) return;                 // wave-uniform guard (EXEC all-1 in WMMA)

    const int m15  = lane & 15;
    const int half = lane >> 4;               // 0: K+0/K+1, 1: K+2/K+3
    const float* xrow = X + (size_t)(tM * 16 + m15) * Din;

    v8f acc[NT];
#pragma unroll
    for (int t = 0; t < NT; ++t)
        acc[t] = (v8f){0.f, 0.f, 0.f, 0.f, 0.f, 0.f, 0.f, 0.f};

    for (int k0 = 0; k0 < Din; k0 += 4) {
        const int ka = k0 + 2 * half;
        v2f a;
        a.x = xrow[ka];
        a.y = xrow[ka + 1];
        const float* wr0 = W + (size_t)ka * K;
        const float* wr1 = wr0 + K;
#pragma unroll
        for (int t = 0; t < NT; ++t) {
            v2f b;
            b.x = wr0[t * 16 + m15];
            b.y = wr1[t * 16 + m15];
            acc[t] = __builtin_amdgcn_wmma_f32_16x16x4_f32(
                false, a, false, b, (short)0, acc[t], false, false);
        }
    }

    const int mbase = tM * 16 + 8 * half;
#pragma unroll
    for (int t = 0; t < NT; ++t) {
        const int n = t * 16 + m15;
        const float bv = (ACT == 1) ? bias[n] : 0.f;
#pragma unroll
        for (int i = 0; i < 8; ++i) {
            float v = acc[t][i];
            if (ACT == 1) v = tanhf(v + bv);
            C[(size_t)(mbase + i) * K + n] = v;
        }
    }
}

// ---------------------------------------------------------------------------
// out[row[e], :] += val[e] * x[col[e], :]    (one wave per edge, D = 64/128)
// ---------------------------------------------------------------------------
__global__ void spmm_atomic_kernel(const int* __restrict__ row,
                                   const int* __restrict__ col,
                                   const float* __restrict__ val,
                                   const float* __restrict__ x,
                                   float* __restrict__ out,
                                   int nnz, int D) {
    const int gid  = blockIdx.x * blockDim.x + threadIdx.x;
    const int e    = gid >> 5;
    const int lane = gid & 31;
    if (e >= nnz) return;
    const int r = row[e];
    const int c = col[e];
    const float v = val[e];
    const float* xr  = x   + (size_t)c * D;
    float*       dst = out + (size_t)r * D;
    for (int j = lane; j < D; j += 32)
        unsafeAtomicAdd(&dst[j], v * xr[j]);
}

__global__ void relu_kernel(float* __restrict__ p, long n) {
    long i = (long)blockIdx.x * blockDim.x + threadIdx.x;
    if (i < n) p[i] = fmaxf(p[i], 0.f);
}

// w[n] = dot(H[n, 0:J], va)
__global__ void rowdot_kernel(const float* __restrict__ H,
                              const float* __restrict__ va,
                              float* __restrict__ w, int Nn, int J) {
    int n = blockIdx.x * blockDim.x + threadIdx.x;
    if (n >= Nn) return;
    float acc = 0.f;
    for (int j = 0; j < J; ++j) acc += H[(size_t)n * J + j] * va[j];
    w[n] = acc;
}

// out[n,d] = b0*v0[n,d] + b1*v1[n,d],  (b0,b1) = softmax(w0[n], w1[n])
__global__ void combine2_kernel(const float* __restrict__ v0,
                                const float* __restrict__ v1,
                                const float* __restrict__ w0,
                                const float* __restrict__ w1,
                                float* __restrict__ out, int Nn, int D) {
    long t = (long)blockIdx.x * blockDim.x + threadIdx.x;
    if (t >= (long)Nn * D) return;
    int n = (int)(t / D);
    float a = w0[n], b = w1[n];
    float m  = fmaxf(a, b);
    float e0 = __expf(a - m), e1 = __expf(b - m);
    float b0 = e0 / (e0 + e1);
    out[t] = b0 * v0[t] + (1.f - b0) * v1[t];
}

// ---------------------------------------------------------------------------
extern "C" void kernel_launch(void* const* d_in, const int* in_sizes, int n_in,
                              void* d_out, int out_size, void* d_ws, size_t ws_size,
                              hipStream_t stream) {
    const float* ft_d = (const float*)d_in[0];
    const float* ft_p = (const float*)d_in[1];
    const int*   dd_row = (const int*)d_in[2];
    const int*   dd_col = (const int*)d_in[3];
    const float* dd_val = (const float*)d_in[4];
    const int*   dp_row = (const int*)d_in[5];
    const int*   dp_col = (const int*)d_in[6];
    const float* dp_val = (const float*)d_in[7];
    const int*   pd_row = (const int*)d_in[8];
    const int*   pd_col = (const int*)d_in[9];
    const float* pd_val = (const float*)d_in[10];
    const float* W0_dd = (const float*)d_in[11];
    const float* W1_dd = (const float*)d_in[12];
    const float* W0_dp = (const float*)d_in[13];
    const float* W1_dp = (const float*)d_in[14];
    const float* W0_pd = (const float*)d_in[15];
    const float* W1_pd = (const float*)d_in[16];
    const float* Wa0_d = (const float*)d_in[17];
    const float* ba0_d = (const float*)d_in[18];
    const float* va0_d = (const float*)d_in[19];
    const float* Wa1_d = (const float*)d_in[20];
    const float* ba1_d = (const float*)d_in[21];
    const float* va1_d = (const float*)d_in[22];
    // indices 23..28 (Wa0_p.. va1_p) are mathematically unused: softmax over 1 relation == identity

    const int Nn  = in_sizes[0] / 128;   // 50000, divisible by 16
    const int NNZ = in_sizes[2];         // 600000

    float* ws  = (float*)d_ws;
    float* Y   = ws;                      // N*128 : GEMM output / spmm input (reused)
    float* V   = Y   + (size_t)Nn * 128;  // 3*N*128 : v_dd, v_dp, v_pd(=embs1_p); later u_dd,u_dp
    float* V0  = V;
    float* V1  = V   + (size_t)Nn * 128;
    float* V2  = V   + (size_t)2 * Nn * 128;
    float* E1d = V   + (size_t)3 * Nn * 128;  // N*128 : embs1_d
    float* H0  = E1d + (size_t)Nn * 128;      // N*32
    float* H1  = H0  + (size_t)Nn * 32;       // N*32
    float* w0  = H1  + (size_t)Nn * 32;       // N
    float* w1  = w0  + Nn;                    // N
    float* outD = (float*)d_out;              // embs2_d : N*64
    float* outP = outD + (size_t)Nn * 64;     // embs2_p : N*64

    const int rowTiles   = Nn / 16;                 // 3125
    const int gemmBlocks = (rowTiles + 3) / 4;      // 4 waves/block
    const int spmmBlocks = (NNZ * 32 + 255) / 256;
    const long n128 = (long)Nn * 128;
    const long n64  = (long)Nn * 64;

    // ---------------- layer 1 : v_rel = relu(spmm(rel, ft @ W0_rel)) --------
    struct { const int* r; const int* c; const float* v; const float* x; const float* W; float* out; }
    rel1[3] = {
        { dd_row, dd_col, dd_val, ft_d, W0_dd, V0 },
        { dp_row, dp_col, dp_val, ft_p, W0_dp, V1 },
        { pd_row, pd_col, pd_val, ft_d, W0_pd, V2 },
    };
    for (int i = 0; i < 3; ++i) {
        gemm_wmma_kernel<8, 0><<<gemmBlocks, 128, 0, stream>>>(
            rel1[i].x, rel1[i].W, nullptr, Y, Nn, 128);
        hipMemsetAsync(rel1[i].out, 0, (size_t)Nn * 128 * sizeof(float), stream);
        spmm_atomic_kernel<<<spmmBlocks, 256, 0, stream>>>(
            rel1[i].r, rel1[i].c, rel1[i].v, Y, rel1[i].out, NNZ, 128);
        relu_kernel<<<(int)((n128 + 255) / 256), 256, 0, stream>>>(rel1[i].out, n128);
    }

    // ---------------- semantic attention (d, layer 1) -----------------------
    gemm_wmma_kernel<2, 1><<<gemmBlocks, 128, 0, stream>>>(V0, Wa0_d, ba0_d, H0, Nn, 128);
    gemm_wmma_kernel<2, 1><<<gemmBlocks, 128, 0, stream>>>(V1, Wa0_d, ba0_d, H1, Nn, 128);
    rowdot_kernel<<<(Nn + 255) / 256, 256, 0, stream>>>(H0, va0_d, w0, Nn, 32);
    rowdot_kernel<<<(Nn + 255) / 256, 256, 0, stream>>>(H1, va0_d, w1, Nn, 32);
    combine2_kernel<<<(int)((n128 + 255) / 256), 256, 0, stream>>>(V0, V1, w0, w1, E1d, Nn, 128);
    // embs1_p == V2 (softmax over a single relation is identity)

    // ---------------- layer 2 : u_rel = relu(spmm(rel, embs1 @ W1_rel)) -----
    float* U0 = V;                        // v_dd/v_dp regions are dead now
    float* U1 = V + (size_t)Nn * 64;
    struct { const int* r; const int* c; const float* v; const float* x; const float* W; float* out; }
    rel2[3] = {
        { dd_row, dd_col, dd_val, E1d, W1_dd, U0 },
        { dp_row, dp_col, dp_val, V2,  W1_dp, U1 },
        { pd_row, pd_col, pd_val, E1d, W1_pd, outP },
    };
    for (int i = 0; i < 3; ++i) {
        gemm_wmma_kernel<4, 0><<<gemmBlocks, 128, 0, stream>>>(
            rel2[i].x, rel2[i].W, nullptr, Y, Nn, 128);
        hipMemsetAsync(rel2[i].out, 0, (size_t)Nn * 64 * sizeof(float), stream);
        spmm_atomic_kernel<<<spmmBlocks, 256, 0, stream>>>(
            rel2[i].r, rel2[i].c, rel2[i].v, Y, rel2[i].out, NNZ, 64);
        relu_kernel<<<(int)((n64 + 255) / 256), 256, 0, stream>>>(rel2[i].out, n64);
    }

    // ---------------- semantic attention (d, layer 2) -----------------------
    gemm_wmma_kernel<1, 1><<<gemmBlocks, 128, 0, stream>>>(U0, Wa1_d, ba1_d, H0, Nn, 64);
    gemm_wmma_kernel<1, 1><<<gemmBlocks, 128, 0, stream>>>(U1, Wa1_d, ba1_d, H1, Nn, 64);
    rowdot_kernel<<<(Nn + 255) / 256, 256, 0, stream>>>(H0, va1_d, w0, Nn, 16);
    rowdot_kernel<<<(Nn + 255) / 256, 256, 0, stream>>>(H1, va1_d, w1, Nn, 16);
    combine2_kernel<<<(int)((n64 + 255) / 256), 256, 0, stream>>>(U0, U1, w0, w1, outD, Nn, 64);
    // embs2_p == outP already written (identity attention over single relation)
}